// cos_loss_32641751449778
// MI455X (gfx1250) — compile-verified
//
#include <hip/hip_runtime.h>
#include <math.h>

// grouped dot(9) -> acos -> global mean.  Memory-bound streaming kernel:
//   283 MB @ 23.3 TB/s  ~= 12 us floor.  NT loads (single-touch data > 192MB L2),
//   LDS regroup for the stride-9 segments, WMMA f32 16x16x4 wave reduction.

typedef __attribute__((ext_vector_type(2))) float v2f;
typedef __attribute__((ext_vector_type(4))) float v4f;
typedef __attribute__((ext_vector_type(8))) float v8f;

#define TPB 256
#define GROUPS_PER_TILE 1024              // 1024 groups * 9 floats = 9216 floats / tile
#define FLOATS_PER_TILE (GROUPS_PER_TILE * 9)

// Reduce the 64 per-lane accumulators of one wave32 with a single
// V_WMMA_F32_16X16X4_F32 against an all-ones B matrix (ones are layout
// invariant, so no B packing needed).  A layout (ISA 7.12.2): lane L<16 holds
// A[L, {0,1}], lane L>=16 holds A[L-16, {2,3}].  D[m,n] = row-sum of A.
// D layout: lane n, vgpr r -> D[r + (lane>=16 ? 8 : 0), n].  Summing the 8
// result VGPRs gives rows 0-7 (lanes 0-15) / rows 8-15 (lanes 16-31); one
// xor-16 shuffle finishes the wave total.  EXEC must be all ones: every
// thread of the block calls this.
__device__ __forceinline__ float wave_reduce_wmma(float a0, float a1) {
    v2f a; a.x = a0; a.y = a1;
    v2f b; b.x = 1.0f; b.y = 1.0f;
    v8f c = {};
    c = __builtin_amdgcn_wmma_f32_16x16x4_f32(
        /*neg_a=*/false, a, /*neg_b=*/false, b,
        /*c_mod=*/(short)0, c, /*reuse_a=*/false, /*reuse_b=*/false);
    float s = c[0] + c[1] + c[2] + c[3] + c[4] + c[5] + c[6] + c[7];
    s += __shfl_xor(s, 16, 32);           // lanes 0-15 hold rows 0-7, 16-31 rows 8-15
    return s;                              // wave total, uniform across lanes
}

__global__ __launch_bounds__(TPB) void cos_loss_stage1(
    const float* __restrict__ x, const float* __restrict__ y,
    float* __restrict__ partials, long long ngroups) {
    __shared__ float prod[FLOATS_PER_TILE];   // 36 KB
    __shared__ float wsum[TPB / 32];

    const int tid = threadIdx.x;
    const long long gbase = (long long)blockIdx.x * GROUPS_PER_TILE;
    long long ng = ngroups - gbase;
    if (ng > GROUPS_PER_TILE) ng = GROUPS_PER_TILE;

    float acc0 = 0.0f, acc1 = 0.0f;

    if (ng == GROUPS_PER_TILE) {
        // ---- fast path: full tile, aligned float4 NT streaming loads ----
        const v4f* __restrict__ xv = (const v4f*)(x + gbase * 9);  // tile base is 16B aligned
        const v4f* __restrict__ yv = (const v4f*)(y + gbase * 9);
        v4f* pv = (v4f*)prod;
#pragma unroll
        for (int k = 0; k < 9; ++k) {
            const int i = tid + k * TPB;                  // 0..2303, fully coalesced
            v4f a = __builtin_nontemporal_load(&xv[i]);   // TH=NT: single-touch stream
            v4f b = __builtin_nontemporal_load(&yv[i]);
            pv[i] = a * b;
        }
        __syncthreads();
        // stride-9 group reads: 9 coprime with 64 banks -> conflict free
#pragma unroll
        for (int k = 0; k < 4; ++k) {
            const int g = tid + k * TPB;
            const float* p = &prod[g * 9];
            float dot = ((p[0] + p[1]) + (p[2] + p[3])) +
                        ((p[4] + p[5]) + (p[6] + p[7])) + p[8];
            float ang = fabsf(acosf((dot - 1.0f) * 0.5f));
            if (k & 1) acc1 += ang; else acc0 += ang;
        }
    } else {
        // ---- tail path (unused for 512x512x15 but keeps the kernel shape-safe) ----
        for (long long g = gbase + tid; g < gbase + ng; g += TPB) {
            const float* xp = x + g * 9;
            const float* yp = y + g * 9;
            float dot = 0.0f;
#pragma unroll
            for (int e = 0; e < 9; ++e) dot += xp[e] * yp[e];
            acc0 += fabsf(acosf((dot - 1.0f) * 0.5f));
        }
    }

    // All 256 threads reach here (EXEC all ones) -> WMMA wave reduction.
    float s = wave_reduce_wmma(acc0, acc1);
    if ((tid & 31) == 0) wsum[tid >> 5] = s;
    __syncthreads();
    if (tid == 0) {
        float t = 0.0f;
#pragma unroll
        for (int w = 0; w < TPB / 32; ++w) t += wsum[w];
        partials[blockIdx.x] = t;          // fixed order -> deterministic
    }
}

__global__ __launch_bounds__(TPB) void cos_loss_stage2(
    const float* __restrict__ partials, int nparts,
    float* __restrict__ out, float inv_count) {
    __shared__ float wsum[TPB / 32];
    const int tid = threadIdx.x;
    float acc0 = 0.0f, acc1 = 0.0f;
    for (int i = tid; i < nparts; i += 2 * TPB) acc0 += partials[i];
    for (int i = tid + TPB; i < nparts; i += 2 * TPB) acc1 += partials[i];

    float s = wave_reduce_wmma(acc0, acc1);
    if ((tid & 31) == 0) wsum[tid >> 5] = s;
    __syncthreads();
    if (tid == 0) {
        float t = 0.0f;
#pragma unroll
        for (int w = 0; w < TPB / 32; ++w) t += wsum[w];
        out[0] = t * inv_count;
    }
}

extern "C" void kernel_launch(void* const* d_in, const int* in_sizes, int n_in,
                              void* d_out, int out_size, void* d_ws, size_t ws_size,
                              hipStream_t stream) {
    const float* x = (const float*)d_in[0];
    const float* y = (const float*)d_in[1];
    const long long total = (long long)in_sizes[0];      // N*T*J*E
    const long long ngroups = total / 9;                  // N*T*J = 3,932,160
    const int ntiles = (int)((ngroups + GROUPS_PER_TILE - 1) / GROUPS_PER_TILE);  // 3840

    float* partials = (float*)d_ws;                       // 3840 * 4 B = 15 KB of scratch

    cos_loss_stage1<<<ntiles, TPB, 0, stream>>>(x, y, partials, ngroups);
    cos_loss_stage2<<<1, TPB, 0, stream>>>(partials, ntiles, (float*)d_out,
                                           1.0f / (float)ngroups);
}